// ACELoss3D_12146167513486
// MI455X (gfx1250) — compile-verified
//
#include <hip/hip_runtime.h>
#include <hip/hip_bf16.h>

#define MIU_C   1.0f
#define EPS_C   1e-8f
// ALPHA * N = 0.001 * (2*3*128*128*128), added once at finalize
#define ALPHA_N (0.001f * 12582912.0f)

typedef __attribute__((ext_vector_type(2))) float v2f;
typedef __attribute__((ext_vector_type(4))) float f4;
typedef __attribute__((ext_vector_type(8))) float v8f;

// Wave32-wide sum via V_WMMA_F32_16X16X4_F32 (ones-B trick, fp32 exact):
// D[m][n] = p_m + p_{m+16}; in-lane sum of 8 D VGPRs collapses 8 rows,
// one xor-16 swap finishes the 32-lane sum.
__device__ __forceinline__ float wave_sum_wmma(float p) {
    v2f a; a.x = p;    a.y = 0.0f;
    v2f b; b.x = 1.0f; b.y = 1.0f;
    v8f c = {0.f, 0.f, 0.f, 0.f, 0.f, 0.f, 0.f, 0.f};
    c = __builtin_amdgcn_wmma_f32_16x16x4_f32(false, a, false, b,
                                              (short)0, c, false, false);
    float s = c[0] + c[1] + c[2] + c[3] + c[4] + c[5] + c[6] + c[7];
    s += __shfl_xor(s, 16, 32);
    return s;
}

#define NSLOT 256  // partial-sum slots in d_ws: NSLOT * 3 floats

__global__ void ace_init(float* acc) {
    for (int i = threadIdx.x; i < NSLOT * 3; i += blockDim.x) acc[i] = 0.0f;
}

// One wave = one (b,c,d,h) row of 128 elements; 4 outputs per lane.
// 256-thread blocks => 8 rows per block; 98304/8 = 12288 blocks.
__global__ __launch_bounds__(256) void ace_main(const float* __restrict__ up,
                                                const float* __restrict__ yt,
                                                float* __restrict__ acc) {
    const int lane = threadIdx.x & 31;
    const int row  = __builtin_amdgcn_readfirstlane(
                         (int)(blockIdx.x * 8u + (threadIdx.x >> 5)));
    const int h  = row & 127;
    const int d  = (row >> 7) & 127;
    const int bc = row >> 14;
    const int hp = (h < 127) ? h + 1 : 127;
    const int hm = (h > 0)   ? h - 1 : 0;
    const int dp = (d < 127) ? d + 1 : 127;
    const int dm = (d > 0)   ? d - 1 : 0;

    const size_t base = (size_t)bc << 21;
    const float* __restrict__ rC  = up + base + ((size_t)d  << 14) + ((size_t)h  << 7);
    const float* __restrict__ rIP = up + base + ((size_t)dp << 14) + ((size_t)h  << 7);
    const float* __restrict__ rIM = up + base + ((size_t)dm << 14) + ((size_t)h  << 7);
    const float* __restrict__ rJP = up + base + ((size_t)d  << 14) + ((size_t)hp << 7);
    const float* __restrict__ rJM = up + base + ((size_t)d  << 14) + ((size_t)hm << 7);
    const float* __restrict__ rPP = up + base + ((size_t)dp << 14) + ((size_t)hp << 7);
    const float* __restrict__ rPM = up + base + ((size_t)dp << 14) + ((size_t)hm << 7);
    const float* __restrict__ rMP = up + base + ((size_t)dm << 14) + ((size_t)hp << 7);
    const float* __restrict__ rMM = up + base + ((size_t)dm << 14) + ((size_t)hm << 7);
    const float* __restrict__ rT  = yt + base + ((size_t)d  << 14) + ((size_t)h  << 7);

    const int k4 = 4 * lane; // lane covers k = k4..k4+3, aligned 16B loads
    const f4 vC  = *(const f4*)(rC  + k4);
    const f4 vIP = *(const f4*)(rIP + k4);
    const f4 vIM = *(const f4*)(rIM + k4);
    const f4 vJP = *(const f4*)(rJP + k4);
    const f4 vJM = *(const f4*)(rJM + k4);
    const f4 vPP = *(const f4*)(rPP + k4);
    const f4 vPM = *(const f4*)(rPM + k4);
    const f4 vMP = *(const f4*)(rMP + k4);
    const f4 vMM = *(const f4*)(rMM + k4);
    const f4 vT  = *(const f4*)(rT  + k4);

    // k-halos via lane shuffles; lane 0/31 clamp to own edge (JAX one-sided form)
    const int lm = lane - 1, lp = lane + 1;
#define EDGE_L(v) ((lane == 0)  ? (v).x : __shfl((v).w, lm, 32))
#define EDGE_R(v) ((lane == 31) ? (v).w : __shfl((v).x, lp, 32))
    const float Ca[6]  = {EDGE_L(vC),  vC.x,  vC.y,  vC.z,  vC.w,  EDGE_R(vC)};
    const float IPa[6] = {EDGE_L(vIP), vIP.x, vIP.y, vIP.z, vIP.w, EDGE_R(vIP)};
    const float IMa[6] = {EDGE_L(vIM), vIM.x, vIM.y, vIM.z, vIM.w, EDGE_R(vIM)};
    const float JPa[6] = {EDGE_L(vJP), vJP.x, vJP.y, vJP.z, vJP.w, EDGE_R(vJP)};
    const float JMa[6] = {EDGE_L(vJM), vJM.x, vJM.y, vJM.z, vJM.w, EDGE_R(vJM)};
#undef EDGE_L
#undef EDGE_R
    const float PPa[4] = {vPP.x, vPP.y, vPP.z, vPP.w};
    const float PMa[4] = {vPM.x, vPM.y, vPM.z, vPM.w};
    const float MPa[4] = {vMP.x, vMP.y, vMP.z, vMP.w};
    const float MMa[4] = {vMM.x, vMM.y, vMM.z, vMM.w};
    const float Ta[4]  = {vT.x,  vT.y,  vT.z,  vT.w};

    float a_in = 0.0f, a_out = 0.0f, a_el = 0.0f;
#pragma unroll
    for (int e = 0; e < 4; ++e) {
        const float uc  = Ca[e + 1];
        const float ci  = 0.5f * (IPa[e + 1] - IMa[e + 1]);
        const float cj  = 0.5f * (JPa[e + 1] - JMa[e + 1]);
        const float ck  = 0.5f * (Ca[e + 2]  - Ca[e]);
        const float cii = IPa[e + 1] + IMa[e + 1] - 2.0f * uc;
        const float cjj = JPa[e + 1] + JMa[e + 1] - 2.0f * uc;
        const float ckk = Ca[e + 2]  + Ca[e]      - 2.0f * uc;
        const float cij = 0.5f * ((PPa[e] - MPa[e]) - (PMa[e] - MMa[e]));
        const float cik = 0.5f * ((IPa[e + 2] - IMa[e + 2]) - (IPa[e] - IMa[e]));
        const float cjk = 0.5f * ((JPa[e + 2] - JMa[e + 2]) - (JPa[e] - JMa[e]));

        const float ci2 = ci * ci, cj2 = cj * cj, ck2 = ck * ck;
        const float s    = ci2 + cj2 + ck2;
        const float len  = __builtin_amdgcn_sqrtf(EPS_C + s);
        const float onep = 1.0f + s;
        float curv = (onep - ck2) * ckk + (onep - ci2) * cii + (onep - cj2) * cjj
                   - 2.0f * cik * cjk * cij;
        curv = fabsf(curv) *
               __builtin_amdgcn_rcpf(__builtin_amdgcn_sqrtf(onep) + EPS_C);
        a_el = fmaf(curv * curv, len, a_el); // ALPHA added analytically at finalize

        const float tt = Ta[e], tm1 = tt - 1.0f;
        a_in  = fmaf(uc, tm1 * tm1, a_in);
        a_out = fmaf(1.0f - uc, tt * tt, a_out);
    }

    // Wave sums through the matrix pipe (all lanes active -> EXEC all-ones)
    const float w_in  = wave_sum_wmma(a_in);
    const float w_out = wave_sum_wmma(a_out);
    const float w_el  = wave_sum_wmma(a_el);

    // Cross-wave combine: plain LDS stores (no atomics -> no readlane loop)
    __shared__ float wsum[8][3];
    if ((threadIdx.x & 31) == 0) {
        const int w = threadIdx.x >> 5;
        wsum[w][0] = w_in;
        wsum[w][1] = w_out;
        wsum[w][2] = w_el;
    }
    __syncthreads();
    if (threadIdx.x == 0) {
        float s0 = 0.f, s1 = 0.f, s2 = 0.f;
#pragma unroll
        for (int w = 0; w < 8; ++w) {
            s0 += wsum[w][0];
            s1 += wsum[w][1];
            s2 += wsum[w][2];
        }
        // scatter across NSLOT slots to avoid same-address atomic serialization
        float* slot = acc + 3u * (blockIdx.x & (NSLOT - 1));
        atomicAdd(&slot[0], s0);
        atomicAdd(&slot[1], s1);
        atomicAdd(&slot[2], s2);
    }
}

// Reduce NSLOT slots (256 threads = 8 waves), combine terms, write scalar.
__global__ __launch_bounds__(NSLOT) void ace_final(const float* __restrict__ acc,
                                                   float* __restrict__ out) {
    const float p_in  = acc[3 * threadIdx.x + 0];
    const float p_out = acc[3 * threadIdx.x + 1];
    const float p_el  = acc[3 * threadIdx.x + 2];
    const float w_in  = wave_sum_wmma(p_in);
    const float w_out = wave_sum_wmma(p_out);
    const float w_el  = wave_sum_wmma(p_el);

    __shared__ float wsum[8][3];
    if ((threadIdx.x & 31) == 0) {
        const int w = threadIdx.x >> 5;
        wsum[w][0] = w_in;
        wsum[w][1] = w_out;
        wsum[w][2] = w_el;
    }
    __syncthreads();
    if (threadIdx.x == 0) {
        float s0 = 0.f, s1 = 0.f, s2 = 0.f;
#pragma unroll
        for (int w = 0; w < 8; ++w) {
            s0 += wsum[w][0];
            s1 += wsum[w][1];
            s2 += wsum[w][2];
        }
        out[0] = MIU_C * fabsf(s0) + fabsf(s1) + (s2 + ALPHA_N);
    }
}

extern "C" void kernel_launch(void* const* d_in, const int* in_sizes, int n_in,
                              void* d_out, int out_size, void* d_ws, size_t ws_size,
                              hipStream_t stream) {
    const float* y_pred = (const float*)d_in[0];
    const float* y_true = (const float*)d_in[1];
    float* out = (float*)d_out;
    float* acc = (float*)d_ws; // NSLOT*3 floats = 3 KB

    ace_init<<<1, 256, 0, stream>>>(acc);

    const int rows = 2 * 3 * 128 * 128;     // 98304 rows
    const int blocks = rows / 8;            // 8 rows (waves) per 256-thread block
    ace_main<<<blocks, 256, 0, stream>>>(y_pred, y_true, acc);

    ace_final<<<1, NSLOT, 0, stream>>>(acc, out);
}